// HOGLayerMoreComplicated_3702261809587
// MI455X (gfx1250) — compile-verified
//
#include <hip/hip_runtime.h>
#include <math.h>

#define NBINS 10
#define HH 512
#define WW 512
#define BB 16
#define CC 3
#define PHD 64              // pooled H
#define PWD 64              // pooled W
#define STRIPS_PER_WG 4
#define WGS_PER_PLANE (PHD / STRIPS_PER_WG)   // 16

typedef unsigned int u32x4 __attribute__((ext_vector_type(4)));
typedef int          i32x4 __attribute__((ext_vector_type(4)));
typedef int          i32x8 __attribute__((ext_vector_type(8)));

// Low 32 bits of a flat (generic) shared-memory pointer are the LDS byte
// offset on gfx1250.
__device__ __forceinline__ unsigned lds_off(const void* p) {
    return (unsigned)(unsigned long long)p;
}

// LDS layout: [2 buffers][8 wave regions][10 rows][64 cols] floats.
// Column c of the 512-wide strip lives in region c>>6 at (c&63).
__device__ __forceinline__ float ldsval(const float* Sb, int lr, int c) {
    return Sb[(((c >> 6) * 10) + lr) * 64 + (c & 63)];
}

__device__ __forceinline__ void load4(const float* Sb, int lr, int c0, float* w) {
    w[0] = (c0 > 0)      ? ldsval(Sb, lr, c0 - 1) : 0.0f;  // zero pad left
    w[1] = ldsval(Sb, lr, c0);
    w[2] = ldsval(Sb, lr, c0 + 1);
    w[3] = (c0 + 2 < WW) ? ldsval(Sb, lr, c0 + 2) : 0.0f;  // zero pad right
}

__device__ __forceinline__ void accum(float gx, float gy, float* hist) {
    float mag  = sqrtf(gx * gx + gy * gy);
    float ph   = atan2f(gx, gy);                 // reference: arctan2(gx, gy)
    float pint = ph * 3.18309886183790672e+00f;  // *10/pi, range (-10, 10]
    int ifl = (int)floorf(pint);                 // [-10, 10]
    int ice = (int)ceilf(pint);                  // [-9, 10]
    if (ifl < 0)      ifl += NBINS;              // Python-style mod -> [0,9]
    if (ifl >= NBINS) ifl -= NBINS;
    if (ice < 0)      ice += NBINS;
    if (ice >= NBINS) ice -= NBINS;
    float onem = 1.0f - mag;
#pragma unroll
    for (int k = 0; k < NBINS; ++k) {
        float t = (k == ifl) ? mag : 0.0f;
        t += (k == ice) ? onem : 0.0f;
        hist[k] += t;
    }
}

// Issue one TDM descriptor: this wave's 64-column x (9|10)-row tile of strip
// `pr` into its LDS region. tensor_dim1 = remaining rows makes the TDM
// zero-fill the bottom halo row (OOB reads return zero).
__device__ __forceinline__ void tdm_issue_slice(const float* xplane, float* ldsbase,
                                                int pr, int waveid) {
    int g0row = pr * 8 - 1;            // top halo row
    int rows  = 10;
    int slot0 = 0;
    if (pr == 0) { g0row = 0; rows = 9; slot0 = 1; }   // row -1 zero-filled manually
    unsigned tdim1 = (unsigned)(HH - g0row);           // valid rows from start
    unsigned long long gaddr =
        (unsigned long long)(xplane + ((size_t)g0row * WW + (size_t)waveid * 64));
    unsigned lds = lds_off(ldsbase + (waveid * 10 + slot0) * 64);

#if __has_builtin(__builtin_amdgcn_tensor_load_to_lds)
    u32x4 g0;
    g0.x = 1u;                                              // count=1 (valid D#)
    g0.y = lds;                                             // lds_addr [63:32]
    g0.z = (unsigned)gaddr;                                 // global_addr [95:64]
    g0.w = (unsigned)((gaddr >> 32) & 0x1FFFFFFull)         // global_addr [120:96]
         | (2u << 30);                                      // type=2 ("image")
    i32x8 g1;
    g1[0] = 0x00020000;                  // wg_mask=0, data_size=2 (4 bytes)
    g1[1] = (int)(512u << 16);           // tensor_dim0 = 512 (bits 79:48)
    g1[2] = (int)((tdim1 & 0xFFFFu) << 16); // tensor_dim1 (bits 111:80)
    g1[3] = (int)(64u << 16);            // tile_dim0 = 64 (bits 127:112)
    g1[4] = rows;                        // tile_dim1 (bits 143:128), tile_dim2=0
    g1[5] = 512;                         // tensor_dim0_stride (bits 207:160)
    g1[6] = 0;                           // stride hi / tensor_dim1_stride
    g1[7] = 0;
    i32x4 z4 = {0, 0, 0, 0};             // groups 2/3 unused (2-D tile)
#if defined(__clang_major__) && (__clang_major__ >= 23)
    i32x8 z8 = {0, 0, 0, 0, 0, 0, 0, 0};
    __builtin_amdgcn_tensor_load_to_lds(g0, g1, z4, z4, z8, 0);
#else
    __builtin_amdgcn_tensor_load_to_lds(g0, g1, z4, z4, 0);
#endif
#else
    // Fallback: per-lane async DMA (32 x b128 = one row-slice per 16 lanes)
    int lane = (int)(threadIdx.x & 31);
#pragma unroll
    for (int k = 0; k < 5; ++k) {
        int j   = lane + k * 32;           // 0..159 transfers of 16B
        int row = j >> 4;                  // 0..9
        int col = (j & 15) << 2;           // 0..60 within this wave's 64 cols
        int r   = row - slot0;
        if (r >= 0 && (unsigned)(g0row + r) < (unsigned)HH) {
            unsigned laddr = lds_off(ldsbase + ((waveid * 10 + row) * 64 + col));
            unsigned voff  = (unsigned)(((g0row + r) * WW + waveid * 64 + col) * 4);
            unsigned long long gb = (unsigned long long)xplane;
            asm volatile("global_load_async_to_lds_b128 %0, %1, %2"
                         :: "v"(laddr), "v"(voff), "s"(gb) : "memory");
        }
    }
    asm volatile("s_wait_asynccnt 0" ::: "memory");
#endif
}

__global__ __launch_bounds__(256) void hog_fused_kernel(
        const float* __restrict__ x, float* __restrict__ out) {
    // Double-buffered strips: 2 x 8 regions x 10 rows x 64 cols = 40 KB LDS
    __shared__ float smem[2][8][10][64];

    const int tid   = threadIdx.x;
    const int wg    = blockIdx.x;
    const int plane = wg / WGS_PER_PLANE;      // 0..47  (b*3 + c)
    const int prb   = wg % WGS_PER_PLANE;      // 0..15  row-block
    const int b     = plane / CC;
    const int c     = plane % CC;
    const float* xplane = x + (size_t)plane * (HH * WW);
    const int waveid = __builtin_amdgcn_readfirstlane((int)(tid >> 5)); // 0..7

    auto issue_strip = [&](int si, int bs) {
        const int pr = prb * STRIPS_PER_WG + si;
        tdm_issue_slice(xplane, &smem[bs][0][0][0], pr, waveid);
        if (pr == 0) {
            // zero-fill the missing top-halo row slot (plain DS store, DScnt)
            int lane = tid & 31;
            *(float2*)&smem[bs][waveid][0][lane * 2] = make_float2(0.f, 0.f);
        }
    };

    const int q  = tid >> 2;        // pooled column 0..63
    const int s  = tid & 3;         // sub-lane within pooled cell
    const int c0 = q * 8 + s * 2;   // this thread owns cols c0, c0+1

    issue_strip(0, 0);

    for (int si = 0; si < STRIPS_PER_WG; ++si) {
        const int bs = si & 1;
        if (si + 1 < STRIPS_PER_WG) {
            issue_strip(si + 1, bs ^ 1);
            // TDM ops complete in order per wave: <=1 outstanding means this
            // wave's slice of strip si has landed in LDS.
#if __has_builtin(__builtin_amdgcn_s_wait_tensorcnt)
            __builtin_amdgcn_s_wait_tensorcnt(1);
        } else {
            __builtin_amdgcn_s_wait_tensorcnt(0);
        }
#else
            asm volatile("s_wait_tensorcnt 1" ::: "memory");
        } else {
            asm volatile("s_wait_tensorcnt 0" ::: "memory");
        }
#endif
        __syncthreads();   // all waves' slices (and zero pads) visible

        // ---- compute strip si from LDS ----------------------------------
        const float* Sb = &smem[bs][0][0][0];
        float hist[NBINS];
#pragma unroll
        for (int k = 0; k < NBINS; ++k) hist[k] = 0.0f;
        float xsum = 0.0f;

        float ra[4], rb[4];
        load4(Sb, 0, c0, ra);
        load4(Sb, 1, c0, rb);
#pragma unroll
        for (int lr = 2; lr < 10; ++lr) {
            float rc[4];
            load4(Sb, lr, c0, rc);
            // center row = lr-1 (strip rows 1..8 = global rows pr*8 .. pr*8+7)
            xsum += rb[1] + rb[2];
            // col c0   : cols (c0-1, c0, c0+1) = indices 0,1,2
            float gx0 = (ra[0] + 2.0f * rb[0] + rc[0]) - (ra[2] + 2.0f * rb[2] + rc[2]);
            float gy0 = (ra[0] + 2.0f * ra[1] + ra[2]) - (rc[0] + 2.0f * rc[1] + rc[2]);
            accum(gx0, gy0, hist);
            // col c0+1 : cols (c0, c0+1, c0+2) = indices 1,2,3
            float gx1 = (ra[1] + 2.0f * rb[1] + rc[1]) - (ra[3] + 2.0f * rb[3] + rc[3]);
            float gy1 = (ra[1] + 2.0f * ra[2] + ra[3]) - (rc[1] + 2.0f * rc[2] + rc[3]);
            accum(gx1, gy1, hist);
#pragma unroll
            for (int t = 0; t < 4; ++t) { ra[t] = rb[t]; rb[t] = rc[t]; }
        }

        // ---- reduce the 4 lanes of each pooled cell (wave32 shuffles) ----
#pragma unroll
        for (int k = 0; k < NBINS; ++k) {
            hist[k] += __shfl_xor(hist[k], 1, 32);
            hist[k] += __shfl_xor(hist[k], 2, 32);
        }
        xsum += __shfl_xor(xsum, 1, 32);
        xsum += __shfl_xor(xsum, 2, 32);

        if (s == 0) {
            const float inv = 1.0f / 64.0f;   // 8x8 mean pool
            const int pr = prb * STRIPS_PER_WG + si;
#pragma unroll
            for (int k = 0; k < NBINS; ++k) {
                size_t o = (((size_t)b * 33 + c * NBINS + k) * PHD + pr) * PWD + q;
                out[o] = hist[k] * inv;
            }
            size_t o = (((size_t)b * 33 + 30 + c) * PHD + pr) * PWD + q;
            out[o] = xsum * inv;
        }
        __syncthreads();   // protect buffer bs before it is re-filled at si+2
    }
}

extern "C" void kernel_launch(void* const* d_in, const int* in_sizes, int n_in,
                              void* d_out, int out_size, void* d_ws, size_t ws_size,
                              hipStream_t stream) {
    (void)in_sizes; (void)n_in; (void)d_ws; (void)ws_size; (void)out_size;
    const float* x = (const float*)d_in[0];   // (16,3,512,512) fp32
    // d_in[1] (Sobel weight) is a compile-time constant in the reference; baked in.
    float* out = (float*)d_out;               // (16,33,64,64) fp32
    dim3 grid(BB * CC * WGS_PER_PLANE);       // 768 workgroups
    dim3 block(256);                          // 8 wave32s
    hog_fused_kernel<<<grid, block, 0, stream>>>(x, out);
}